// Attention_59296318488685
// MI455X (gfx1250) — compile-verified
//
#include <hip/hip_runtime.h>
#include <hip/hip_bf16.h>

#define B_  32
#define S_  2048
#define H_  1024
#define SZ_ 1024
#define D_  1024
#define NCHUNK 16

typedef __attribute__((ext_vector_type(2))) float v2f;
typedef __attribute__((ext_vector_type(8))) float v8f;

// ---------------------------------------------------------------------------
// Generic skinny WMMA GEMM: C = [tanh](A @ B), one 16x16 tile per wave.
// A element (m,k) at A + m*ldaM + k          (k contiguous)
// B element (k,n) at B + k*ldbK + n*ldbN     (ldbK=ldb,ldbN=1 normal; ldbK=1,ldbN=ldb transposed)
// Uses V_WMMA_F32_16X16X4_F32: A 16x4 (2 VGPR), B 4x16 (2 VGPR), C 16x16 f32 (8 VGPR).
// ---------------------------------------------------------------------------
__global__ void wmma_gemm_tanh(const float* __restrict__ A, long ldaM,
                               const float* __restrict__ Bm, long ldbK, long ldbN,
                               float* __restrict__ Cm, int ldc,
                               int Mtiles, int Ntiles, int K, int apply_tanh)
{
    int waveInBlock = threadIdx.x >> 5;
    int lane        = threadIdx.x & 31;
    int tile        = blockIdx.x * (blockDim.x >> 5) + waveInBlock;
    if (tile >= Mtiles * Ntiles) return;              // wave-uniform: EXEC stays all-1s
    int mt = tile / Ntiles;
    int nt = tile % Ntiles;

    int half = lane >> 4;     // 0: K=0,1   1: K=2,3   (A and B symmetric per ISA layout)
    int idx  = lane & 15;     // A: M index   B: N index

    const float* Arow = A  + (long)(mt * 16 + idx) * ldaM;
    const float* Bcol = Bm + (long)(nt * 16 + idx) * ldbN;

    v8f c = {0.f, 0.f, 0.f, 0.f, 0.f, 0.f, 0.f, 0.f};
    for (int k0 = 0; k0 < K; k0 += 4) {
        int k = k0 + half * 2;
        v2f a, b;
        a.x = Arow[k];
        a.y = Arow[k + 1];
        b.x = Bcol[(long)k * ldbK];
        b.y = Bcol[(long)(k + 1) * ldbK];
        c = __builtin_amdgcn_wmma_f32_16x16x4_f32(false, a, false, b,
                                                  (short)0, c, false, false);
    }

    // C/D layout: VGPR r -> lanes 0-15: M=r, lanes 16-31: M=r+8 ; N = lane&15
    int n     = nt * 16 + idx;
    int mbase = mt * 16 + half * 8;
    #pragma unroll
    for (int r = 0; r < 8; ++r) {
        float v = c[r];
        if (apply_tanh) v = tanhf(v);
        Cm[(long)(mbase + r) * ldc + n] = v;
    }
}

// aligned[b] = sigmoid(P[b,:] . v_p) * S
__global__ void aligned_kernel(const float* __restrict__ P,
                               const float* __restrict__ v_p,
                               float* __restrict__ alignedv)
{
    __shared__ float red[8];
    int b = blockIdx.x;
    float acc = 0.f;
    for (int i = threadIdx.x; i < H_; i += 256)
        acc += P[(long)b * H_ + i] * v_p[i];
    #pragma unroll
    for (int o = 16; o >= 1; o >>= 1) acc += __shfl_xor(acc, o, 32);
    if ((threadIdx.x & 31) == 0) red[threadIdx.x >> 5] = acc;
    __syncthreads();
    if (threadIdx.x == 0) {
        float s = 0.f;
        #pragma unroll
        for (int i = 0; i < 8; ++i) s += red[i];
        alignedv[b] = (1.f / (1.f + expf(-s))) * (float)S_;
    }
}

// scores[b,s] = x[b,s,:] . U[b,:]   — one wave per row, float4 streaming
__global__ void scores_kernel(const float* __restrict__ x,
                              const float* __restrict__ U,
                              float* __restrict__ scores)
{
    int gw   = (blockIdx.x * blockDim.x + threadIdx.x) >> 5;   // row index b*S+s
    int lane = threadIdx.x & 31;
    int b    = gw >> 11;                                       // / S_
    const float* xr = x + (long)gw * H_;
    const float* ur = U + (long)b  * H_;
    float acc = 0.f;
    #pragma unroll
    for (int i = 0; i < H_ / (32 * 4); ++i) {
        int off = (lane + i * 32) * 4;
        float4 xv = *(const float4*)(xr + off);
        float4 uv = *(const float4*)(ur + off);
        acc += xv.x * uv.x + xv.y * uv.y + xv.z * uv.z + xv.w * uv.w;
    }
    #pragma unroll
    for (int o = 16; o >= 1; o >>= 1) acc += __shfl_xor(acc, o, 32);
    if (lane == 0) scores[gw] = acc;
}

// attn[b,s] = softmax_s(scores[b,:]) * exp(-(s-aligned[b])^2 / (2*(D/2)^2))
__global__ void softmax_gauss_kernel(const float* __restrict__ scores,
                                     const float* __restrict__ alignedv,
                                     float* __restrict__ attn)
{
    __shared__ float redmax[8];
    __shared__ float redsum[8];
    int b = blockIdx.x;
    int t = threadIdx.x;
    const float* srow = scores + (long)b * S_;

    float vals[8];
    float m = -1e30f;
    #pragma unroll
    for (int i = 0; i < 8; ++i) {
        vals[i] = srow[t + i * 256];
        m = fmaxf(m, vals[i]);
    }
    #pragma unroll
    for (int o = 16; o >= 1; o >>= 1) m = fmaxf(m, __shfl_xor(m, o, 32));
    if ((t & 31) == 0) redmax[t >> 5] = m;
    __syncthreads();
    float bm = redmax[0];
    #pragma unroll
    for (int i = 1; i < 8; ++i) bm = fmaxf(bm, redmax[i]);

    float s = 0.f;
    #pragma unroll
    for (int i = 0; i < 8; ++i) {
        vals[i] = expf(vals[i] - bm);
        s += vals[i];
    }
    #pragma unroll
    for (int o = 16; o >= 1; o >>= 1) s += __shfl_xor(s, o, 32);
    if ((t & 31) == 0) redsum[t >> 5] = s;
    __syncthreads();
    float total = 0.f;
    #pragma unroll
    for (int i = 0; i < 8; ++i) total += redsum[i];
    float inv = 1.f / total;

    float al = alignedv[b];
    const float inv_ss = 1.0f / (2.0f * (D_ * 0.5f) * (D_ * 0.5f));
    #pragma unroll
    for (int i = 0; i < 8; ++i) {
        int pos = t + i * 256;
        float d = (float)pos - al;
        attn[(long)b * S_ + pos] = vals[i] * inv * expf(-d * d * inv_ss);
    }
}

// partial[b,chunk,h] = sum over chunk's s of attn[b,s] * x[b,s,h]
__global__ void context_partial_kernel(const float* __restrict__ x,
                                       const float* __restrict__ attn,
                                       float* __restrict__ partial)
{
    __shared__ float aw[S_ / NCHUNK];                 // 128 weights
    int b = blockIdx.y, chunk = blockIdx.x, t = threadIdx.x;
    int s0 = chunk * (S_ / NCHUNK);
    if (t < S_ / NCHUNK) aw[t] = attn[(long)b * S_ + s0 + t];
    __syncthreads();

    const float* xb = x + ((long)b * S_ + s0) * H_ + t * 4;
    float4 acc = make_float4(0.f, 0.f, 0.f, 0.f);
    for (int s = 0; s < S_ / NCHUNK; ++s) {
        float a = aw[s];
        float4 xv = *(const float4*)(xb + (long)s * H_);
        acc.x += a * xv.x; acc.y += a * xv.y;
        acc.z += a * xv.z; acc.w += a * xv.w;
    }
    *(float4*)(partial + (long)(b * NCHUNK + chunk) * H_ + t * 4) = acc;
}

// combined[b, 0:H]   = sum_chunks partial   (context)
// combined[b, H:2H]  = h_t[b] = x[b, S-1, :]
__global__ void build_combined_kernel(const float* __restrict__ partial,
                                      const float* __restrict__ x,
                                      float* __restrict__ combined)
{
    int b = blockIdx.x;
    for (int h = threadIdx.x; h < H_; h += blockDim.x) {
        float s = 0.f;
        #pragma unroll
        for (int c = 0; c < NCHUNK; ++c)
            s += partial[(long)(b * NCHUNK + c) * H_ + h];
        combined[(long)b * 2 * H_ + h]      = s;
        combined[(long)b * 2 * H_ + H_ + h] = x[((long)b * S_ + (S_ - 1)) * H_ + h];
    }
}

extern "C" void kernel_launch(void* const* d_in, const int* in_sizes, int n_in,
                              void* d_out, int out_size, void* d_ws, size_t ws_size,
                              hipStream_t stream)
{
    const float* x   = (const float*)d_in[0];   // (B,S,H)
    const float* W_p = (const float*)d_in[1];   // (H,H)
    const float* v_p = (const float*)d_in[2];   // (H,1)
    const float* W_a = (const float*)d_in[3];   // (H,H)
    const float* W_v = (const float*)d_in[4];   // (2H,SIZE)
    float* out = (float*)d_out;                 // (B,SIZE)

    // workspace layout (floats), ~3.1 MB total
    float* ws       = (float*)d_ws;
    float* P        = ws;                               // B*H
    float* U        = P + B_ * H_;                      // B*H
    float* alignedv = U + B_ * H_;                      // 64 (padded)
    float* scores   = alignedv + 64;                    // B*S
    float* attn     = scores + (long)B_ * S_;           // B*S
    float* partial  = attn + (long)B_ * S_;             // B*NCHUNK*H
    float* combined = partial + (long)B_ * NCHUNK * H_; // B*2H

    const float* HT = x + (long)(S_ - 1) * H_;          // h_t rows, stride S*H

    // P = tanh(HT @ W_p)         M=32,N=1024,K=1024   (128 tiles, 8 waves/block)
    wmma_gemm_tanh<<<16, 256, 0, stream>>>(HT, (long)S_ * H_, W_p, (long)H_, 1L,
                                           P, H_, 2, 64, H_, 1);
    // U = HT @ W_a^T             (B strides swapped -> transposed access)
    wmma_gemm_tanh<<<16, 256, 0, stream>>>(HT, (long)S_ * H_, W_a, 1L, (long)H_,
                                           U, H_, 2, 64, H_, 0);
    // aligned positions
    aligned_kernel<<<B_, 256, 0, stream>>>(P, v_p, alignedv);
    // scores: streaming pass 1 over x (one wave per (b,s))
    scores_kernel<<<(B_ * S_ * 32) / 256, 256, 0, stream>>>(x, U, scores);
    // softmax * gaussian window
    softmax_gauss_kernel<<<B_, 256, 0, stream>>>(scores, alignedv, attn);
    // context: streaming pass 2 over x
    dim3 g5(NCHUNK, B_);
    context_partial_kernel<<<g5, 256, 0, stream>>>(x, attn, partial);
    build_combined_kernel<<<B_, 256, 0, stream>>>(partial, x, combined);
    // out = tanh(combined @ W_v)   M=32,N=1024,K=2048
    wmma_gemm_tanh<<<16, 256, 0, stream>>>(combined, 2L * H_, W_v, (long)SZ_, 1L,
                                           out, SZ_, 2, 64, 2 * H_, 1);
}